// ModelRNN_13589276524881
// MI455X (gfx1250) — compile-verified
//
#include <hip/hip_runtime.h>

typedef __attribute__((ext_vector_type(2))) float v2f;
typedef __attribute__((ext_vector_type(8))) float v8f;

#define B_ 128
#define S_ 256
#define C_ 512
#define H_ 512
#define A_ 512
#define V_ 256
#define T_ 128
#define GATES_ (4 * H_)
#define XDIM_ (V_ + A_) /* 768 */

// ---------------------------------------------------------------------------
// zero-init kernel (h, c must be zeroed every call; harness does not re-poison)
// ---------------------------------------------------------------------------
__global__ void zero_kernel(float* __restrict__ p, int n) {
  int i = blockIdx.x * blockDim.x + threadIdx.x;
  if (i < n) p[i] = 0.0f;
}

// ---------------------------------------------------------------------------
// Generic fp32 WMMA GEMM, NN layout:
//   D[M,N] = A[M,K](row-major, lda) * B[K,N](row-major, ldb) + bias[N]
// One wave computes a 16x64 strip of D (A fragment reused across 4 N-tiles).
// Fragment layouts per CDNA5 ISA 7.12.2 (32-bit A 16x4, 32-bit C/D 16x16).
// M % 16 == 0, N % 64 == 0, K % 4 == 0 required (true for all calls here).
// ---------------------------------------------------------------------------
__global__ void gemm_wmma_nn(const float* __restrict__ A, int lda,
                             const float* __restrict__ Bm, int ldb,
                             float* __restrict__ D, int ldd,
                             const float* __restrict__ bias,
                             int M, int N, int K) {
  const int wave = (blockIdx.x * blockDim.x + threadIdx.x) >> 5;
  const int lane = threadIdx.x & 31;
  const int n_strips = N >> 6;
  const int m_tiles = M >> 4;
  if (wave >= m_tiles * n_strips) return; // wave-uniform: EXEC stays all-1s
  const int m_tile = wave / n_strips;
  const int n_strip = wave - m_tile * n_strips;
  const int hl = lane >> 4;   // half-select: K-pair for A/B, M+8 for C/D
  const int l16 = lane & 15;
  const int row = m_tile * 16 + l16;
  const float* __restrict__ Arow = A + (size_t)row * lda;
  const int n0 = n_strip * 64 + l16;

  v8f acc[4] = {};
  for (int k0 = 0; k0 < K; k0 += 4) {
    const int ka = k0 + 2 * hl;
    v2f a;
    a.x = Arow[ka];
    a.y = Arow[ka + 1];
#pragma unroll
    for (int j = 0; j < 4; ++j) {
      const int col = n0 + j * 16;
      v2f b;
      b.x = Bm[(size_t)ka * ldb + col];
      b.y = Bm[(size_t)(ka + 1) * ldb + col];
      acc[j] = __builtin_amdgcn_wmma_f32_16x16x4_f32(
          false, a, false, b, (short)0, acc[j], false, false);
    }
  }
#pragma unroll
  for (int j = 0; j < 4; ++j) {
    const int col = n0 + j * 16;
    const float bv = bias ? bias[col] : 0.0f;
#pragma unroll
    for (int r = 0; r < 8; ++r) {
      const int orow = m_tile * 16 + r + 8 * hl;
      D[(size_t)orow * ldd + col] = acc[j][r] + bv;
    }
  }
}

// ---------------------------------------------------------------------------
// Fused gates GEMM:
//   gates[128,2048] = ctx @ W_ih[:,V:].T + h @ W_hh.T
//                     + W_ih[:, lab_b]   (one-hot input -> column gather)
//                     + b_ih + b_hh
// Both weight factors are NT (B element (k,n) = W[n*ldK + k]).
// ---------------------------------------------------------------------------
__global__ void gates_wmma_kernel(const float* __restrict__ ctx,
                                  const float* __restrict__ h,
                                  const float* __restrict__ W_ih,
                                  const float* __restrict__ W_hh,
                                  const float* __restrict__ b_ih,
                                  const float* __restrict__ b_hh,
                                  const int* __restrict__ labels, int tcol,
                                  float* __restrict__ gates) {
  const int wave = (blockIdx.x * blockDim.x + threadIdx.x) >> 5;
  const int lane = threadIdx.x & 31;
  const int n_strips = GATES_ >> 6; // 32
  if (wave >= (B_ >> 4) * n_strips) return;
  const int m_tile = wave / n_strips;
  const int n_strip = wave - m_tile * n_strips;
  const int hl = lane >> 4;
  const int l16 = lane & 15;
  const int row = m_tile * 16 + l16;
  const float* __restrict__ ctx_row = ctx + (size_t)row * A_;
  const float* __restrict__ h_row = h + (size_t)row * H_;
  const int n0 = n_strip * 64 + l16;

  v8f acc[4] = {};
  // ctx @ W_ih[:, V:].T
  for (int k0 = 0; k0 < A_; k0 += 4) {
    const int ka = k0 + 2 * hl;
    v2f a;
    a.x = ctx_row[ka];
    a.y = ctx_row[ka + 1];
#pragma unroll
    for (int j = 0; j < 4; ++j) {
      const int col = n0 + j * 16;
      const float* bp = W_ih + (size_t)col * XDIM_ + V_ + ka;
      v2f b;
      b.x = bp[0];
      b.y = bp[1];
      acc[j] = __builtin_amdgcn_wmma_f32_16x16x4_f32(
          false, a, false, b, (short)0, acc[j], false, false);
    }
  }
  // h @ W_hh.T
  for (int k0 = 0; k0 < H_; k0 += 4) {
    const int ka = k0 + 2 * hl;
    v2f a;
    a.x = h_row[ka];
    a.y = h_row[ka + 1];
#pragma unroll
    for (int j = 0; j < 4; ++j) {
      const int col = n0 + j * 16;
      const float* bp = W_hh + (size_t)col * H_ + ka;
      v2f b;
      b.x = bp[0];
      b.y = bp[1];
      acc[j] = __builtin_amdgcn_wmma_f32_16x16x4_f32(
          false, a, false, b, (short)0, acc[j], false, false);
    }
  }
  // epilogue: biases + one-hot column gather
#pragma unroll
  for (int j = 0; j < 4; ++j) {
    const int col = n0 + j * 16;
    const float bv = b_ih[col] + b_hh[col];
#pragma unroll
    for (int r = 0; r < 8; ++r) {
      const int orow = m_tile * 16 + r + 8 * hl; // batch index
      const int lab = labels[orow * T_ + tcol];
      gates[(size_t)orow * GATES_ + col] =
          acc[j][r] + bv + W_ih[(size_t)col * XDIM_ + lab];
    }
  }
}

// ---------------------------------------------------------------------------
// Attention: one block (256 threads) per batch row.
//   scores[s] = dot(q[b], keys[b,s,:]) / sqrt(A); softmax; ctx = w @ keys[b]
// ---------------------------------------------------------------------------
__global__ void attn_kernel(const float* __restrict__ q,
                            const float* __restrict__ keys,
                            float* __restrict__ ctx) {
  __shared__ float qs[A_];   // 512
  __shared__ float red[S_];  // 256 (reductions)
  __shared__ float w[S_];    // 256 (softmax weights)
  const int b = blockIdx.x;
  const int tid = threadIdx.x; // 0..255
  const float* __restrict__ qb = q + (size_t)b * A_;
  qs[tid] = qb[tid];
  qs[tid + 256] = qb[tid + 256];
  __syncthreads();

  // one score per thread
  const float* __restrict__ krow = keys + ((size_t)b * S_ + tid) * A_;
  float sc = 0.0f;
  for (int k = 0; k < A_; k += 4) {
    sc += krow[k] * qs[k] + krow[k + 1] * qs[k + 1] +
          krow[k + 2] * qs[k + 2] + krow[k + 3] * qs[k + 3];
  }
  sc *= 0.044194173824159216f; // 1/sqrt(512)

  // block max
  red[tid] = sc;
  __syncthreads();
  for (int s = 128; s > 0; s >>= 1) {
    if (tid < s) red[tid] = fmaxf(red[tid], red[tid + s]);
    __syncthreads();
  }
  const float mx = red[0];
  __syncthreads();
  // block sum of exp
  const float e = expf(sc - mx);
  red[tid] = e;
  __syncthreads();
  for (int s = 128; s > 0; s >>= 1) {
    if (tid < s) red[tid] += red[tid + s];
    __syncthreads();
  }
  const float inv = 1.0f / red[0];
  w[tid] = e * inv;
  __syncthreads();

  // ctx[b, a] for a = tid and tid+256 (coalesced over keys rows)
  const float* __restrict__ kb = keys + (size_t)b * S_ * A_;
  float acc0 = 0.0f, acc1 = 0.0f;
  for (int s = 0; s < S_; ++s) {
    const float ws = w[s];
    const float* kr = kb + (size_t)s * A_;
    acc0 += ws * kr[tid];
    acc1 += ws * kr[tid + 256];
  }
  ctx[(size_t)b * A_ + tid] = acc0;
  ctx[(size_t)b * A_ + tid + 256] = acc1;
}

// ---------------------------------------------------------------------------
// LSTM pointwise: gate order (i, f, g, o); updates h, c in place.
// ---------------------------------------------------------------------------
__global__ void lstm_kernel(const float* __restrict__ gates,
                            float* __restrict__ h, float* __restrict__ c) {
  const int idx = blockIdx.x * blockDim.x + threadIdx.x;
  if (idx >= B_ * H_) return;
  const int b = idx >> 9;        // / 512
  const int j = idx & (H_ - 1);  // % 512
  const float* g = gates + (size_t)b * GATES_;
  const float gi = g[j];
  const float gf = g[H_ + j];
  const float gg = g[2 * H_ + j];
  const float go = g[3 * H_ + j];
  const float si = 1.0f / (1.0f + expf(-gi));
  const float sf = 1.0f / (1.0f + expf(-gf));
  const float so = 1.0f / (1.0f + expf(-go));
  const float cn = sf * c[idx] + si * tanhf(gg);
  const float hn = so * tanhf(cn);
  c[idx] = cn;
  h[idx] = hn;
}

// ---------------------------------------------------------------------------
// Host launch: keys GEMM once, then T=128 dependent steps on one stream
// (graph-captured once by the harness, so launch count is cheap on replay).
// ---------------------------------------------------------------------------
extern "C" void kernel_launch(void* const* d_in, const int* in_sizes, int n_in,
                              void* d_out, int out_size, void* d_ws,
                              size_t ws_size, hipStream_t stream) {
  (void)in_sizes; (void)n_in; (void)out_size; (void)ws_size;
  const float* embed = (const float*)d_in[0];  // [B,S,C]
  const int* labels = (const int*)d_in[1];     // [B,T]
  const float* W_Ic = (const float*)d_in[2];   // [C,A]
  const float* b_Ic = (const float*)d_in[3];   // [A]
  const float* W_Hc = (const float*)d_in[4];   // [H,A]
  const float* b_Hc = (const float*)d_in[5];   // [A]
  const float* W_ih = (const float*)d_in[6];   // [4H, V+A]
  const float* b_ih = (const float*)d_in[7];   // [4H]
  const float* W_hh = (const float*)d_in[8];   // [4H, H]
  const float* b_hh = (const float*)d_in[9];   // [4H]
  const float* W_out = (const float*)d_in[10]; // [H,V]
  const float* b_out = (const float*)d_in[11]; // [V]
  float* out = (float*)d_out;                  // [B,T,V]

  float* ws = (float*)d_ws;
  float* keys = ws;                                // B*S*A = 16,777,216 f
  float* h = keys + (size_t)B_ * S_ * A_;          // 65,536 f
  float* c = h + B_ * H_;                          // 65,536 f
  float* q = c + B_ * H_;                          // 65,536 f
  float* ctx = q + B_ * A_;                        // 65,536 f
  float* gates = ctx + B_ * A_;                    // 262,144 f

  // h = c = 0
  zero_kernel<<<(2 * B_ * H_ + 255) / 256, 256, 0, stream>>>(h, 2 * B_ * H_);

  // keys = embed @ W_Ic + b_Ic   (M=32768, N=512, K=512)
  {
    const int M = B_ * S_, N = A_, K = C_;
    const int blocks = ((M >> 4) * (N >> 6) * 32 + 255) / 256; // 2048
    gemm_wmma_nn<<<blocks, 256, 0, stream>>>(embed, C_, W_Ic, A_, keys, A_,
                                             b_Ic, M, N, K);
  }

  for (int t = 0; t < T_; ++t) {
    // q = h @ W_Hc + b_Hc   (128 x 512, K=512)
    gemm_wmma_nn<<<8, 256, 0, stream>>>(h, H_, W_Hc, A_, q, A_, b_Hc,
                                        B_, A_, H_);
    // attention -> ctx
    attn_kernel<<<B_, 256, 0, stream>>>(q, keys, ctx);
    // gates (uses pre-update h; teacher-forced input = label at t-1, t0 uses 0)
    const int tcol = (t == 0) ? 0 : (t - 1);
    gates_wmma_kernel<<<32, 256, 0, stream>>>(ctx, h, W_ih, W_hh, b_ih, b_hh,
                                              labels, tcol, gates);
    // pointwise LSTM -> h, c (in place)
    lstm_kernel<<<(B_ * H_ + 255) / 256, 256, 0, stream>>>(gates, h, c);
    // out[:, t, :] = h_new @ W_out + b_out  (128 x 256, K=512)
    gemm_wmma_nn<<<4, 256, 0, stream>>>(h, H_, W_out, V_, out + (size_t)t * V_,
                                        T_ * V_, b_out, B_, V_, H_);
  }
}